// MambaBlock_87041807220760
// MI455X (gfx1250) — compile-verified
//
#include <hip/hip_runtime.h>
#include <hip/hip_bf16.h>

// ---------------------------------------------------------------------------
// Mamba block forward for MI455X (gfx1250, CDNA5, wave32).
//
// ~27 GFLOP of GEMM vs ~150 MB traffic -> memory/latency bound, so:
//  * V_WMMA_F32_16X16X4_F32 consumes the f32 tensors directly (no convert
//    passes, full precision); bf16/fp8 peak is irrelevant at this intensity.
//  * Tiles staged global->LDS with GLOBAL_LOAD_ASYNC_TO_LDS_B128 (ASYNCcnt),
//    double-buffered, so WMMA feeds from LDS instead of eating VMEM latency
//    per fragment (the round-0 kernel serialized load->wait->wmma).
// ---------------------------------------------------------------------------

typedef __attribute__((ext_vector_type(2))) float v2f;
typedef __attribute__((ext_vector_type(8))) float v8f;

#define L_SEQ    2048
#define DMODEL   1024
#define DINNER   2048
#define DTRANK   64
#define DSTATE   16
#define DCONV    4

#define BK   32      // K-chunk staged per LDS buffer
#define BKP  36      // padded LDS row stride (floats): 36r mod 64 distinct for r=0..15

__device__ __forceinline__ float softplus_f(float x) {
    return (x > 20.f) ? x : __logf(1.f + __expf(x));
}
__device__ __forceinline__ float sigmoid_f(float x) {
    return 1.f / (1.f + __expf(-x));
}

// LDS byte offset of a shared-memory object (generic -> addrspace(3) cast).
__device__ __forceinline__ unsigned lds_off(const void* p) {
    return (unsigned)(uintptr_t)(__attribute__((address_space(3))) const void*)p;
}

// one async 16-byte global->LDS copy (per lane); tracked by ASYNCcnt
__device__ __forceinline__ void async_copy16(unsigned lds_byte, const float* gaddr) {
    asm volatile("global_load_async_to_lds_b128 %0, %1, off"
                 :: "v"(lds_byte), "v"(gaddr) : "memory");
}

// ---------------------------------------------------------------------------
// LDS-staged NT-form WMMA GEMM:  C[M,N] = epi( A[M,K] * B[N,K]^T )
// Block = 4 waves: 64 x (16*NT) C tile; wave w owns rows [16w,16w+16).
// Fragment layout per CDNA5 ISA (32-bit, 16x16x4):
//   lane&15 = m/n index, lane>>4 = K half; A/B frag = float2 at k + 2*khalf.
//   C frag (float8): element v -> row (khalf*8 + v), col (lane&15).
// EPI: 0 none | 1 +bias[n], softplus | 2 +addsrc[m*ldc+n] (residual)
// ---------------------------------------------------------------------------
template <int NT, int EPI>
__global__ __launch_bounds__(128)
void wmma_gemm_lds(const float* __restrict__ A,
                   const float* __restrict__ B,
                   float* __restrict__ C,
                   const float* __restrict__ bias,
                   const float* __restrict__ addsrc,
                   int M, int N, int K,
                   int lda, int ldb, int ldc)
{
    __shared__ __align__(16) float As[2][64 * BKP];
    __shared__ __align__(16) float Bs[2][16 * NT * BKP];

    const int lane  = threadIdx.x;                // 0..31
    const int wv    = threadIdx.y;                // 0..3
    const int tid   = wv * 32 + lane;             // 0..127
    const int mrow  = lane & 15;
    const int khalf = lane >> 4;
    const int tileM0 = blockIdx.y * 64;
    const int tileN0 = blockIdx.x * (16 * NT);

    // stage one K-chunk (A: 64x32, B: 16*NT x 32) into LDS buffer `buf`;
    // each wave issues (4 + NT) async B128 copies.
    auto stage = [&](int kc0, int buf) {
#pragma unroll
        for (int j = 0; j < 4; ++j) {
            const int f = tid + j * 128;          // float4 index in A tile
            const int row = f >> 3, k4 = f & 7;
            async_copy16(lds_off(&As[buf][row * BKP + k4 * 4]),
                         A + (size_t)(tileM0 + row) * lda + kc0 + k4 * 4);
        }
#pragma unroll
        for (int j = 0; j < NT; ++j) {
            const int f = tid + j * 128;          // float4 index in B tile
            const int row = f >> 3, k4 = f & 7;
            async_copy16(lds_off(&Bs[buf][row * BKP + k4 * 4]),
                         B + (size_t)(tileN0 + row) * ldb + kc0 + k4 * 4);
        }
    };

    v8f acc[NT] = {};

    stage(0, 0);
    const int nch = K / BK;
    for (int ch = 0; ch < nch; ++ch) {
        const int cur = ch & 1;
        if (ch + 1 < nch) {
            stage((ch + 1) * BK, cur ^ 1);
            // async ops complete in order: <= (4+NT) outstanding means the
            // older chunk's (4+NT) copies have landed in LDS.
            asm volatile("s_wait_asynccnt %0" :: "i"(4 + NT) : "memory");
        } else {
            asm volatile("s_wait_asynccnt 0x0" ::: "memory");
        }
        __syncthreads();

        const float* as = &As[cur][(wv * 16 + mrow) * BKP + 2 * khalf];
        const float* bs = &Bs[cur][mrow * BKP + 2 * khalf];
#pragma unroll
        for (int kk = 0; kk < BK; kk += 4) {
            v2f a = *(const v2f*)(as + kk);
#pragma unroll
            for (int t = 0; t < NT; ++t) {
                v2f b = *(const v2f*)(bs + t * 16 * BKP + kk);
                acc[t] = __builtin_amdgcn_wmma_f32_16x16x4_f32(
                    false, a, false, b, (short)0, acc[t], false, false);
            }
        }
        __syncthreads();
    }

#pragma unroll
    for (int t = 0; t < NT; ++t) {
        const int col = tileN0 + t * 16 + mrow;
        const float bv = (EPI == 1) ? bias[col] : 0.f;
#pragma unroll
        for (int v = 0; v < 8; ++v) {
            const int row = tileM0 + wv * 16 + khalf * 8 + v;
            float val = acc[t][v];
            if (EPI == 1) val = softplus_f(val + bv);
            if (EPI == 2) val += addsrc[(size_t)row * ldc + col];
            C[(size_t)row * ldc + col] = val;
        }
    }
}

// ---------------------------------------------------------------------------
// LayerNorm: one block per token row of 1024.
// ---------------------------------------------------------------------------
__global__ void layernorm_k(const float* __restrict__ x,
                            const float* __restrict__ gamma,
                            const float* __restrict__ beta,
                            float* __restrict__ xn)
{
    const int l = blockIdx.x;
    const float* row = x + (size_t)l * DMODEL;
    float s = 0.f, s2 = 0.f;
    for (int i = threadIdx.x; i < DMODEL; i += 256) {
        float v = row[i];
        s += v; s2 += v * v;
    }
#pragma unroll
    for (int off = 16; off > 0; off >>= 1) {
        s  += __shfl_down(s,  off, 32);
        s2 += __shfl_down(s2, off, 32);
    }
    __shared__ float red[2][8];
    const int wave = threadIdx.x >> 5;
    if ((threadIdx.x & 31) == 0) { red[0][wave] = s; red[1][wave] = s2; }
    __syncthreads();
    if (threadIdx.x == 0) {
        float a = 0.f, b = 0.f;
#pragma unroll
        for (int w = 0; w < 8; ++w) { a += red[0][w]; b += red[1][w]; }
        red[0][0] = a; red[1][0] = b;
    }
    __syncthreads();
    const float mean = red[0][0] * (1.f / DMODEL);
    const float var  = red[1][0] * (1.f / DMODEL) - mean * mean;
    const float rstd = rsqrtf(var + 1e-5f);
    for (int i = threadIdx.x; i < DMODEL; i += 256) {
        xn[(size_t)l * DMODEL + i] = (row[i] - mean) * rstd * gamma[i] + beta[i];
    }
}

// ---------------------------------------------------------------------------
// Depthwise causal conv1d (width 4) + bias + SiLU.  x_part is the first
// DINNER columns of xz (row stride 2*DINNER).
// ---------------------------------------------------------------------------
__global__ void conv_silu_k(const float* __restrict__ xz,
                            const float* __restrict__ conv_w,
                            const float* __restrict__ conv_b,
                            float* __restrict__ xc)
{
    const int idx = blockIdx.x * blockDim.x + threadIdx.x;   // l*DINNER + c
    if (idx >= L_SEQ * DINNER) return;
    const int c = idx & (DINNER - 1);
    const int l = idx >> 11;
    const float* w = conv_w + c * DCONV;
    float acc = conv_b[c];
#pragma unroll
    for (int j = 0; j < DCONV; ++j) {
        const int ll = l - (DCONV - 1) + j;
        if (ll >= 0) acc += xz[(size_t)ll * (2 * DINNER) + c] * w[j];
    }
    xc[idx] = acc * sigmoid_f(acc);
}

// ---------------------------------------------------------------------------
// Selective scan: lane = (channel, state); 16 states of a channel occupy a
// contiguous 16-lane group -> 4 shfl_xor hops reduce over states (wave32).
// Fuses  y = (sum_n h*C + u*Dp) * silu(z).
// ---------------------------------------------------------------------------
__global__ void scan_k(const float* __restrict__ delta,
                       const float* __restrict__ xc,
                       const float* __restrict__ xdbl,   // [L,96] = dt|B|C
                       const float* __restrict__ A_log,
                       const float* __restrict__ Dp,
                       const float* __restrict__ xz,     // z at col DINNER..
                       float* __restrict__ y)
{
    const int tid = threadIdx.x;                 // 256 = 16 chan x 16 states
    const int n = tid & 15;
    const int d = blockIdx.x * 16 + (tid >> 4);
    const float Acoef = -__expf(A_log[d * DSTATE + n]);
    const float dp = Dp[d];
    float h = 0.f;
    for (int l = 0; l < L_SEQ; ++l) {
        const float dl = delta[(size_t)l * DINNER + d];
        const float u  = xc[(size_t)l * DINNER + d];
        const float Bv = xdbl[(size_t)l * 96 + DTRANK + n];
        const float Cv = xdbl[(size_t)l * 96 + DTRANK + DSTATE + n];
        h = __expf(dl * Acoef) * h + dl * Bv * u;
        float contrib = h * Cv;
        contrib += __shfl_xor(contrib, 1, 32);
        contrib += __shfl_xor(contrib, 2, 32);
        contrib += __shfl_xor(contrib, 4, 32);
        contrib += __shfl_xor(contrib, 8, 32);
        if (n == 0) {
            const float z = xz[(size_t)l * (2 * DINNER) + DINNER + d];
            y[(size_t)l * DINNER + d] = (contrib + u * dp) * (z * sigmoid_f(z));
        }
    }
}

// ---------------------------------------------------------------------------
extern "C" void kernel_launch(void* const* d_in, const int* in_sizes, int n_in,
                              void* d_out, int out_size, void* d_ws, size_t ws_size,
                              hipStream_t stream)
{
    const float* x      = (const float*)d_in[0];
    const float* gamma  = (const float*)d_in[1];
    const float* beta   = (const float*)d_in[2];
    const float* w_in   = (const float*)d_in[3];
    const float* conv_w = (const float*)d_in[4];
    const float* conv_b = (const float*)d_in[5];
    const float* w_x    = (const float*)d_in[6];
    const float* w_dt   = (const float*)d_in[7];
    const float* b_dt   = (const float*)d_in[8];
    const float* A_log  = (const float*)d_in[9];
    const float* Dp     = (const float*)d_in[10];
    const float* w_out  = (const float*)d_in[11];
    float* out = (float*)d_out;

    // workspace carve-up (floats): ~22.2M floats = ~89 MB
    float* ws    = (float*)d_ws;
    float* xn    = ws;                                   // L*DMODEL
    float* xz    = xn   + (size_t)L_SEQ * DMODEL;        // L*2*DINNER
    float* xc    = xz   + (size_t)L_SEQ * 2 * DINNER;    // L*DINNER
    float* xdbl  = xc   + (size_t)L_SEQ * DINNER;        // L*96
    float* delta = xdbl + (size_t)L_SEQ * 96;            // L*DINNER
    float* ybuf  = delta+ (size_t)L_SEQ * DINNER;        // L*DINNER

    const dim3 blk(32, 4);   // 4 waves per block

    // 1) LayerNorm
    layernorm_k<<<L_SEQ, 256, 0, stream>>>(x, gamma, beta, xn);

    // 2) in-projection: xz[L,4096] = xn[L,1024] * w_in[4096,1024]^T
    wmma_gemm_lds<4, 0><<<dim3(2 * DINNER / 64, L_SEQ / 64), blk, 0, stream>>>(
        xn, w_in, xz, nullptr, nullptr,
        L_SEQ, 2 * DINNER, DMODEL, DMODEL, DMODEL, 2 * DINNER);

    // 3) depthwise conv + SiLU
    conv_silu_k<<<(L_SEQ * DINNER) / 256, 256, 0, stream>>>(xz, conv_w, conv_b, xc);

    // 4) x_dbl[L,96] = xc[L,2048] * w_x[96,2048]^T   (NT=6 covers N=96)
    wmma_gemm_lds<6, 0><<<dim3(1, L_SEQ / 64), blk, 0, stream>>>(
        xc, w_x, xdbl, nullptr, nullptr,
        L_SEQ, 96, DINNER, DINNER, DINNER, 96);

    // 5) delta[L,2048] = softplus(dt[L,64] * w_dt[2048,64]^T + b_dt)
    //    dt = xdbl columns 0..63 (lda = 96); K=64 -> 2 LDS chunks
    wmma_gemm_lds<4, 1><<<dim3(DINNER / 64, L_SEQ / 64), blk, 0, stream>>>(
        xdbl, w_dt, delta, b_dt, nullptr,
        L_SEQ, DINNER, DTRANK, 96, DTRANK, DINNER);

    // 6) selective scan + SiLU(z) gate
    scan_k<<<DINNER / 16, 256, 0, stream>>>(delta, xc, xdbl, A_log, Dp, xz, ybuf);

    // 7) out[L,1024] = ybuf[L,2048] * w_out[1024,2048]^T + residual x
    wmma_gemm_lds<4, 2><<<dim3(DMODEL / 64, L_SEQ / 64), blk, 0, stream>>>(
        ybuf, w_out, out, nullptr, x,
        L_SEQ, DMODEL, DINNER, DINNER, DINNER, DMODEL);
}